// CatalogLensingSystem_14053132992586
// MI455X (gfx1250) — compile-verified
//
#include <hip/hip_runtime.h>
#include <hip/hip_bf16.h>
#include <math.h>

// Fused catalog-lensing render for MI455X (gfx1250, wave32).
//  - TDM (tensor_load_to_lds + s_wait_tensorcnt) stages param tables into LDS
//  - per-block derived constants in LDS (cos/sin/ecc/bfac, gaussian weights)
//  - per-pixel PEMD deflection (transcendental VALU)
//  - gaussian sources via V_WMMA_F32_16X16X4_F32 with A = source-weight rows,
//    B = pixel-feature columns, so D columns land on the owning lane:
//    exponent[src, pix] = [a,-2a x0,-2a y0,a(x0^2+y0^2)] . [sx^2+sy^2, sx, sy, 1]
// Output written exactly once (37.7MB) -> HBM-write/launch bound (~1.6us at
// 23.3 TB/s); fusing kills the 150MB deflection intermediate of the ref graph.

typedef float v2f __attribute__((ext_vector_type(2)));
typedef float v8f __attribute__((ext_vector_type(8)));

namespace {
constexpr int kB  = 64;
constexpr int kH  = 384;
constexpr int kW  = 384;
constexpr int kHW = kH * kW;            // 147456
constexpr int kNP = 96;                 // PEMD components
constexpr int kNS = 96;                 // gaussian sources
constexpr int kThreads = 256;           // 8 waves
constexpr int kPixPerThread = 4;
constexpr int kTilesPerBatch = kHW / (kThreads * kPixPerThread); // 144 exact
}

// ---------------- TDM support (gfx1250 async tensor path) ----------------
#if defined(__has_builtin)
#  if __has_builtin(__builtin_amdgcn_tensor_load_to_lds)
#    define HAVE_TDM 1
#  endif
#  if __has_builtin(__builtin_amdgcn_s_wait_tensorcnt)
#    define HAVE_WAIT_TENSOR 1
#  endif
#endif
#if __has_include(<hip/amd_detail/amd_gfx1250_TDM.h>)
#  define TDM_ARITY6 1   // therock/clang-23 headers -> 6-arg builtin
#endif

#ifdef HAVE_TDM
typedef unsigned u32x4 __attribute__((ext_vector_type(4)));
typedef int      i32x4 __attribute__((ext_vector_type(4)));
typedef int      i32x8 __attribute__((ext_vector_type(8)));

typedef __attribute__((address_space(3))) void lds_void;

__device__ __forceinline__ unsigned lds_offset(const void* p) {
  // generic -> LDS addrspace cast; AS(3) pointer value is the LDS byte offset
  return (unsigned)(unsigned long long)(lds_void*)p;
}

// 1-D tensor DMA: copy ndw dwords global -> LDS (D# per ISA 08 S8.3-8.5).
__device__ __forceinline__ void tdm_load_1d(const float* gsrc, float* ldst, int ndw) {
  const unsigned long long ga = (unsigned long long)gsrc;
  u32x4 g0;
  g0[0] = 1u;                                           // count=1, user-mode, no gather
  g0[1] = lds_offset(ldst);                             // lds_addr (bytes)
  g0[2] = (unsigned)(ga & 0xffffffffu);                 // global_addr[31:0]
  g0[3] = (unsigned)((ga >> 32) & 0x01ffffffu) | (2u << 30);  // addr[56:32] | type=2
  i32x8 g1;
  const unsigned n = (unsigned)ndw;
  g1[0] = 0x20000;                                      // wg_mask=0, data_size=4B
  g1[1] = (int)((n & 0xffffu) << 16);                   // tensor_dim0[15:0]  @ bits 63:48
  g1[2] = (int)((n >> 16) | (1u << 16));                // dim0[31:16] | tensor_dim1=1
  g1[3] = (int)((n & 0xffffu) << 16);                   // dim1 hi=0 | tile_dim0 = ndw
  g1[4] = 1;                                            // tile_dim1=1, tile_dim2=0
  g1[5] = (int)n;                                       // tensor_dim0_stride[31:0]
  g1[6] = 0;                                            // stride0 hi | stride1 lo
  g1[7] = 0;                                            // stride1 hi
  i32x4 z4 = {0, 0, 0, 0};
#ifdef TDM_ARITY6
  i32x8 z8 = {0, 0, 0, 0, 0, 0, 0, 0};
  __builtin_amdgcn_tensor_load_to_lds(g0, g1, z4, z4, z8, 0);
#else
  __builtin_amdgcn_tensor_load_to_lds(g0, g1, z4, z4, 0);
#endif
}

__device__ __forceinline__ void wait_tensorcnt0() {
#ifdef HAVE_WAIT_TENSOR
  __builtin_amdgcn_s_wait_tensorcnt(0);
#else
  asm volatile("s_wait_tensorcnt 0" ::: "memory");
#endif
}
#endif  // HAVE_TDM

__global__ __launch_bounds__(kThreads)
void lens_render_kernel(const float* __restrict__ lens_grid,
                        const float* __restrict__ pemd_params,
                        const float* __restrict__ precomp_params,
                        const float* __restrict__ source_params,
                        const int*   __restrict__ batch_idx,
                        const int*   __restrict__ pemd_sys_idx,
                        const int*   __restrict__ precomp_map,
                        const int*   __restrict__ source_sys_idx,
                        float*       __restrict__ out) {
  __shared__ float sRawP  [kNP * 6];   // staged pemd params
  __shared__ float sRawPre[kNP * 4];   // staged precomp params
  __shared__ float sRawS  [kNS * 4];   // staged source params
  __shared__ float sPemd[kNP * 8];     // derived: x0,y0,cos,sin,q^2,ecc,bfac,pad
  __shared__ float sSrc [kNS * 5];     // derived: a,-2a*x0,-2a*y0,a*(x0^2+y0^2),amp

  const int tid  = threadIdx.x;
  const int lane = tid & 31;
  const int b    = blockIdx.y;

  if (tid == 0) {
    __builtin_prefetch(lens_grid, 0, 3);   // global_prefetch_b8
  }
  const int pmap0 = precomp_map[0];

  // ---- phase 0: stage raw tables into LDS via Tensor Data Mover ----
#ifdef HAVE_TDM
  if (tid < 32) {   // one DMA per descriptor, issued by wave 0 (EXEC ignored)
    tdm_load_1d(pemd_params,    sRawP,   kNP * 6);
    tdm_load_1d(precomp_params, sRawPre, kNP * 4);
    tdm_load_1d(source_params,  sRawS,   kNS * 4);
    wait_tensorcnt0();
  }
#else
  for (int j = tid; j < kNP * 6; j += kThreads) sRawP[j]   = pemd_params[j];
  for (int j = tid; j < kNP * 4; j += kThreads) sRawPre[j] = precomp_params[j];
  for (int j = tid; j < kNS * 4; j += kThreads) sRawS[j]   = source_params[j];
#endif
  __syncthreads();

  // ---- phase 1: derive per-component constants (one thread per component) ----
  if (tid < kNP) {
    const int k = tid;
    const float x0  = sRawP[k * 6 + 0];
    const float y0  = sRawP[k * 6 + 1];
    const float q   = sRawP[k * 6 + 2];
    const float phi = sRawP[k * 6 + 3];
    const float thE = sRawP[k * 6 + 4];
    const float scl = sRawPre[k * 4 + pmap0];
    const float c   = __cosf(phi);
    const float s   = __sinf(phi);
    const float ecc = sqrtf(fmaxf(1.0f - q * q, 1e-8f));
    float* pk = &sPemd[k * 8];
    pk[0] = x0; pk[1] = y0; pk[2] = c; pk[3] = s;
    pk[4] = q * q; pk[5] = ecc; pk[6] = thE * scl * q / ecc; pk[7] = 0.f;
  }
  if (tid >= 128 && tid < 128 + kNS) {
    const int m = tid - 128;
    const float x0 = sRawS[m * 4 + 0];
    const float y0 = sRawS[m * 4 + 1];
    const float sg = sRawS[m * 4 + 2];
    const float am = sRawS[m * 4 + 3];
    const float a  = -0.5f / (sg * sg);
    float* sw = &sSrc[m * 5];
    sw[0] = a;
    sw[1] = -2.0f * a * x0;
    sw[2] = -2.0f * a * y0;
    sw[3] = a * fmaf(x0, x0, y0 * y0);
    sw[4] = am;
  }
  __syncthreads();

  // ---- component / source ranges for this batch slot (sorted sys_idx) ----
  const int bval = batch_idx[b];
  int cLo = 0; while (cLo < kNP && pemd_sys_idx[cLo] <  bval) ++cLo;
  int cHi = cLo; while (cHi < kNP && pemd_sys_idx[cHi] == bval) ++cHi;
  int sLo = 0; while (sLo < kNS && source_sys_idx[sLo] <  bval) ++sLo;
  int sHi = sLo; while (sHi < kNS && source_sys_idx[sHi] == bval) ++sHi;

  const float2* __restrict__ grid2 = (const float2*)lens_grid;
  float* __restrict__ outb = out + (size_t)b * kHW;
  const bool hi = (lane >= 16);

  for (int it = 0; it < kPixPerThread; ++it) {
    const int pix = blockIdx.x * (kThreads * kPixPerThread) + it * kThreads + tid;
    const float2 g = grid2[pix];
    const float gx = g.x, gy = g.y;

    // ---- PEMD deflection accumulation ----
    float axT = 0.f, ayT = 0.f;
    for (int k = cLo; k < cHi; ++k) {
      const float* pk = &sPemd[k * 8];
      const float x0 = pk[0], y0 = pk[1], c = pk[2], s = pk[3];
      const float q2 = pk[4], ecc = pk[5], bfac = pk[6];
      const float x  = gx - x0, y = gy - y0;
      const float xr = fmaf(c, x, s * y);
      const float yr = fmaf(-s, x, c * y);
      const float psi = sqrtf(fmaf(q2 * xr, xr, yr * yr)) + 1e-12f;
      const float inv = 1.0f / psi;
      const float tx = ecc * xr * inv;
      float ty = ecc * yr * inv;
      ty = fminf(fmaxf(ty, -1.0f + 1e-6f), 1.0f - 1e-6f);
      const float axk = bfac * atanf(tx);
      const float ayk = bfac * 0.5f * __logf((1.0f + ty) / (1.0f - ty));  // atanh
      axT = fmaf(c, axk, fmaf(-s, ayk, axT));
      ayT = fmaf(s, axk, fmaf(c, ayk, ayT));
    }
    const float sx = gx - axT;
    const float sy = gy - ayT;

    // ---- gaussian sources via WMMA f32 16x16x4 (A = sources, B = pixels) ----
    // B (4x16): lanes 0-15: v0=K0=sx^2+sy^2, v1=K1=sx of pixel col (lane&15);
    //           lanes 16-31: v0=K2=sy, v1=K3=1.
    const float sxA = __shfl(sx, lane & 15, 32);
    const float syA = __shfl(sy, lane & 15, 32);
    const float sxB = __shfl(sx, 16 + (lane & 15), 32);
    const float syB = __shfl(sy, 16 + (lane & 15), 32);
    v2f B1, B2;
    B1[0] = hi ? syA : fmaf(sxA, sxA, syA * syA);
    B1[1] = hi ? 1.0f : sxA;
    B2[0] = hi ? syB : fmaf(sxB, sxB, syB * syB);
    B2[1] = hi ? 1.0f : sxB;

    float bright = 0.f;
    for (int s0 = sLo; s0 < sHi; s0 += 16) {
      // A (16x4): row m = source s0+m; lanes 0-15 carry K0/K1, 16-31 carry K2/K3
      const int si = s0 + (lane & 15);
      const int sc = (si < sHi) ? si : (sHi - 1);
      const float msk = (si < sHi) ? 1.0f : 0.0f;
      const float* sw = &sSrc[sc * 5];
      v2f Av;
      Av[0] = (hi ? sw[2] : sw[0]) * msk;
      Av[1] = (hi ? sw[3] : sw[1]) * msk;

      // per-row amplitudes for this lane's half (rows 0-7 lo / 8-15 hi)
      float amp[8];
      #pragma unroll
      for (int m = 0; m < 8; ++m) {
        const int sm = s0 + m + (hi ? 8 : 0);
        amp[m] = (sm < sHi) ? sSrc[sm * 5 + 4] : 0.0f;
      }

      v8f cz = {0.f, 0.f, 0.f, 0.f, 0.f, 0.f, 0.f, 0.f};
      v8f d1 = __builtin_amdgcn_wmma_f32_16x16x4_f32(false, Av, false, B1,
                                                     (short)0, cz, false, false);
      v8f d2 = __builtin_amdgcn_wmma_f32_16x16x4_f32(false, Av, false, B2,
                                                     (short)0, cz, false, false);
      // D: lane n holds column/pixel n (n = lane&15), rows (0..7)+8*(lane>>4).
      float p1 = 0.f, p2 = 0.f;
      #pragma unroll
      for (int m = 0; m < 8; ++m) {
        p1 = fmaf(amp[m], __expf(d1[m]), p1);
        p2 = fmaf(amp[m], __expf(d2[m]), p2);
      }
      // combine the two row-halves living in lanes n and n^16
      p1 += __shfl_xor(p1, 16, 32);
      p2 += __shfl_xor(p2, 16, 32);
      bright += hi ? p2 : p1;   // lane's own pixel: pass1 cols 0-15, pass2 cols 16-31
    }

    outb[pix] = bright;   // coalesced; every output element written exactly once
  }
}

extern "C" void kernel_launch(void* const* d_in, const int* in_sizes, int n_in,
                              void* d_out, int out_size, void* d_ws, size_t ws_size,
                              hipStream_t stream) {
  (void)in_sizes; (void)n_in; (void)d_ws; (void)ws_size; (void)out_size;
  const float* lens_grid      = (const float*)d_in[0];
  const float* pemd_params    = (const float*)d_in[1];
  const float* precomp_params = (const float*)d_in[2];
  const float* source_params  = (const float*)d_in[3];
  const int*   batch_idx      = (const int*)d_in[4];
  const int*   pemd_sys_idx   = (const int*)d_in[5];
  // d_in[6] precomp_sys_idx: identical to pemd_sys_idx per reference; unused.
  const int*   precomp_map    = (const int*)d_in[7];
  const int*   source_sys_idx = (const int*)d_in[8];
  float* out = (float*)d_out;

  dim3 grid(kTilesPerBatch, kB);
  dim3 block(kThreads);
  hipLaunchKernelGGL(lens_render_kernel, grid, block, 0, stream,
                     lens_grid, pemd_params, precomp_params, source_params,
                     batch_idx, pemd_sys_idx, precomp_map, source_sys_idx, out);
}